// DynamicControls_74680891343567
// MI455X (gfx1250) — compile-verified
//
#include <hip/hip_runtime.h>
#include <hip/hip_bf16.h>

typedef __attribute__((ext_vector_type(2))) float v2f;
typedef __attribute__((ext_vector_type(8))) float v8f;

#define TT 64
#define NN 64
#define TS 64
#define CC 512
#define NEGL (-1.0e30f)

// ---------- helpers ----------

__device__ __forceinline__ float lse2(float a, float b) {
    float m = fmaxf(a, b);
    float d = fabsf(a - b);
    return m + log1pf(expf(-d));
}

// Inclusive prefix-LSE over 64 elems striped 2-per-lane across a wave32.
// lane l holds states s=2l (x0) and s=2l+1 (x1).
__device__ __forceinline__ void wave_prefix_lse64(float x0, float x1, float& y0, float& y1) {
    int lane = threadIdx.x & 31;
    float pt = lse2(x0, x1);          // pair total
    float p = pt;
#pragma unroll
    for (int d = 1; d < 32; d <<= 1) {
        float o = __shfl_up(p, d, 32);
        if (lane >= d) p = lse2(p, o);
    }
    float e = __shfl_up(p, 1, 32);    // exclusive pair prefix (valid lane>=1)
    if (lane == 0) { y0 = x0; y1 = pt; }
    else           { y0 = lse2(e, x0); y1 = lse2(e, pt); }
}

// Inclusive suffix-LSE over 64 elems, 2-per-lane.
__device__ __forceinline__ void wave_suffix_lse64(float x0, float x1, float& y0, float& y1) {
    int lane = threadIdx.x & 31;
    float pt = lse2(x0, x1);
    float p = pt;
#pragma unroll
    for (int d = 1; d < 32; d <<= 1) {
        float o = __shfl_down(p, d, 32);
        if (lane + d < 32) p = lse2(p, o);
    }
    float e = __shfl_down(p, 1, 32);  // exclusive pair suffix (valid lane<31)
    if (lane == 31) { y1 = x1; y0 = pt; }
    else            { y1 = lse2(e, x1); y0 = lse2(e, pt); }
}

// 64-thread-block inclusive prefix-LSE via LDS; leaves inclusive values in lds.
__device__ __forceinline__ float blk_prefix_lse(float v, float* lds, int s) {
    lds[s] = v; __syncthreads();
#pragma unroll
    for (int d = 1; d < 64; d <<= 1) {
        float o = (s >= d) ? lds[s - d] : NEGL;
        __syncthreads();
        if (s >= d) v = lse2(v, o);
        lds[s] = v; __syncthreads();
    }
    return v;
}

// 64-thread-block inclusive suffix-LSE via LDS; leaves inclusive values in lds.
__device__ __forceinline__ float blk_suffix_lse(float v, float* lds, int s) {
    lds[s] = v; __syncthreads();
#pragma unroll
    for (int d = 1; d < 64; d <<= 1) {
        float o = (s + d < 64) ? lds[s + d] : NEGL;
        __syncthreads();
        if (s + d < 64) v = lse2(v, o);
        lds[s] = v; __syncthreads();
    }
    return v;
}

// ---------- K1: gate GEMV via f32 WMMA (memory-bound: streams 512MB of x) ----------
// One wave handles 16 rows of x (each row = 512 floats); 128 WMMA 16x16x4_f32,
// with B = gate_w broadcast across all 16 columns -> every column of D equals g.
__global__ __launch_bounds__(256) void gate_wmma_kernel(
        const float* __restrict__ x, const float* __restrict__ gw,
        float* __restrict__ g) {
    __shared__ float wl[CC];
    int tid = threadIdx.x;
    wl[tid] = gw[tid];
    wl[tid + 256] = gw[tid + 256];
    __syncthreads();

    int wave = tid >> 5;
    int lane = tid & 31;
    long rowbase = ((long)blockIdx.x * 8 + wave) * 16;
    int row16 = lane & 15;
    int koff  = (lane >> 4) << 1;   // lanes 0-15: K={0,1}; lanes 16-31: K={2,3}
    const float* xp = x + (size_t)(rowbase + row16) * CC + koff;

    v8f acc = {};
#pragma unroll 4
    for (int c = 0; c < CC / 4; ++c) {
        v2f a = *(const v2f*)(xp + 4 * c);
        v2f b;
        b.x = wl[4 * c + koff];
        b.y = wl[4 * c + koff + 1];
        // D = A(16x4) * B(4x16) + C ; all 16 columns of B identical (= gate_w chunk)
        acc = __builtin_amdgcn_wmma_f32_16x16x4_f32(
            false, a, false, b, (short)0, acc, false, false);
    }
    // C/D layout: VGPR r -> M=r (lanes 0-15), M=r+8 (lanes 16-31); all columns equal.
    if (lane == 0) {
#pragma unroll
        for (int r = 0; r < 8; ++r) g[rowbase + r] = acc[r];
    } else if (lane == 16) {
#pragma unroll
        for (int r = 0; r < 8; ++r) g[rowbase + 8 + r] = acc[r];
    }
}

// ---------- K2: log-sigmoid, controls output, exclusive cumsum cs, c1m ----------
__global__ __launch_bounds__(64) void controls_kernel(
        const float* __restrict__ g, const float* __restrict__ gbp,
        float* __restrict__ out_ctrl, float* __restrict__ cs_ws,
        float* __restrict__ c1m_ws) {
    __shared__ float lds[TS];
    int tn = blockIdx.x;            // t*NN + n
    int t = tn >> 6;
    int s = threadIdx.x;
    int idx = tn * TS + s;

    float gv = g[idx] + gbp[0];
    float sv = fminf(gv, 0.0f) - log1pf(expf(-fabsf(gv)));  // log_sigmoid(g)
    float c1 = sv - gv;                                     // log_sigmoid(-g)
    if (t < TT - 1) {
        out_ctrl[idx * 2 + 0] = sv;
        out_ctrl[idx * 2 + 1] = c1;
    }
    c1m_ws[idx] = (s == TS - 1) ? 0.0f : c1;

    // exclusive prefix sum of c0 = sv
    float v = sv;
    lds[s] = v; __syncthreads();
#pragma unroll
    for (int d = 1; d < 64; d <<= 1) {
        float o = (s >= d) ? lds[s - d] : 0.0f;
        __syncthreads();
        if (s >= d) v += o;
        lds[s] = v; __syncthreads();
    }
    cs_ws[idx] = (s == 0) ? 0.0f : lds[s - 1];
}

// ---------- K3: forward scan (serial over t; prefix-LSE per step) ----------
__global__ __launch_bounds__(32) void forward_kernel(
        const float* __restrict__ e, const float* __restrict__ cs_ws,
        const float* __restrict__ c1m_ws, float* __restrict__ alpha_ws,
        float* __restrict__ prior_ws) {
    int n = blockIdx.x;
    int lane = threadIdx.x;
    int s0 = 2 * lane;

    float a0 = (lane == 0) ? e[n * TS] : NEGL;   // alpha0: only state 0 live
    float a1 = NEGL;
    *(float2*)(alpha_ws + n * TS + s0) = make_float2(a0, a1);

    for (int t = 0; t < TT - 1; ++t) {
        int base = (t * NN + n) * TS + s0;
        float2 cs  = *(const float2*)(cs_ws  + base);
        float2 c1m = *(const float2*)(c1m_ws + base);
        int baseN = ((t + 1) * NN + n) * TS + s0;
        float2 ev  = *(const float2*)(e + baseN);

        float L0, L1;
        wave_prefix_lse64(a0 - cs.x, a1 - cs.y, L0, L1);
        a0 = L0 + c1m.x + cs.x + ev.x;
        a1 = L1 + c1m.y + cs.y + ev.y;
        *(float2*)(alpha_ws + baseN) = make_float2(a0, a1);
    }
    // prior = LSE over final alpha
    float pt = lse2(a0, a1);
#pragma unroll
    for (int d = 16; d >= 1; d >>= 1) pt = lse2(pt, __shfl_xor(pt, d, 32));
    if (lane == 0) prior_ws[n] = pt;
}

// ---------- K4: backward scan (serial over t; suffix-LSE per step) ----------
__global__ __launch_bounds__(32) void backward_kernel(
        const float* __restrict__ e, const float* __restrict__ cs_ws,
        const float* __restrict__ c1m_ws, float* __restrict__ beta_ws) {
    int n = blockIdx.x;
    int lane = threadIdx.x;
    int s0 = 2 * lane;

    float b0 = 0.0f, b1 = 0.0f;
    *(float2*)(beta_ws + ((TT - 1) * NN + n) * TS + s0) = make_float2(0.0f, 0.0f);

    for (int t = TT - 2; t >= 0; --t) {
        int base = (t * NN + n) * TS + s0;
        float2 cs  = *(const float2*)(cs_ws  + base);
        float2 c1m = *(const float2*)(c1m_ws + base);
        float2 ev  = *(const float2*)(e + ((t + 1) * NN + n) * TS + s0);

        float S0, S1;
        wave_suffix_lse64(c1m.x + cs.x + b0 + ev.x,
                          c1m.y + cs.y + b1 + ev.y, S0, S1);
        b0 = S0 - cs.x;
        b1 = S1 - cs.y;
        *(float2*)(beta_ws + base) = make_float2(b0, b1);
    }
}

// ---------- K5: gamma / write / read / emissions copy ----------
__global__ __launch_bounds__(64) void finalize_kernel(
        const float* __restrict__ e, const float* __restrict__ alpha_ws,
        const float* __restrict__ beta_ws, const float* __restrict__ prior_ws,
        const float* __restrict__ cs_ws, const float* __restrict__ c1m_ws,
        float* __restrict__ out) {
    __shared__ float lds[TS];
    int tn = blockIdx.x;
    int t = tn >> 6;
    int n = tn & 63;
    int s = threadIdx.x;
    int idx = tn * TS + s;

    float* out_em    = out;                                   // (64,64,64)
    float* out_gamma = out + TT * NN * TS;                    // (64,64,64)
    float* out_read  = out + 2 * TT * NN * TS + (TT - 1) * NN * TS * 2;  // after controls
    float* out_write = out_read + (TT - 1) * NN * TS;

    float pr = prior_ws[n];
    float a = alpha_ws[idx];
    float b = beta_ws[idx];
    float gamma = expf(a + b - pr);

    out_em[idx] = e[idx];
    out_gamma[idx] = gamma;
    if (t >= 1) out_write[((t - 1) * NN + n) * TS + s] = gamma;

    if (t <= TT - 2) {   // uniform per block
        float cs = cs_ws[idx];
        // prefix over p_i = alpha[t,i] - cs[i], inclusive up to s
        float A = blk_prefix_lse(a - cs, lds, s);
        __syncthreads();
        // suffix over q_j = c1m[j]+cs[j]+beta[t+1,j]+e[t+1,j], strictly j > s
        int idxN = ((t + 1) * NN + n) * TS + s;
        float q = c1m_ws[idx] + cs + beta_ws[idxN] + e[idxN];
        blk_suffix_lse(q, lds, s);
        float S = (s == TS - 1) ? NEGL : lds[s + 1];
        out_read[idx] = expf(A + S - pr);
    }
}

// ---------- launch ----------
extern "C" void kernel_launch(void* const* d_in, const int* in_sizes, int n_in,
                              void* d_out, int out_size, void* d_ws, size_t ws_size,
                              hipStream_t stream) {
    (void)in_sizes; (void)n_in; (void)out_size; (void)ws_size;
    const float* x  = (const float*)d_in[0];   // (64,64,64,512)
    const float* em = (const float*)d_in[1];   // (64,64,64)
    const float* gw = (const float*)d_in[2];   // (512,)
    const float* gb = (const float*)d_in[3];   // (1,)
    float* out = (float*)d_out;

    float* W = (float*)d_ws;
    const int P = TT * NN * TS;                // 262144
    float* g_ws     = W;
    float* cs_ws    = W + P;
    float* c1m_ws   = W + 2 * P;
    float* alpha_ws = W + 3 * P;
    float* beta_ws  = W + 4 * P;
    float* prior_ws = W + 5 * P;               // 64 floats

    float* out_ctrl = out + 2 * P;             // controls region (63,64,64,2)

    // K1: 262144 rows / 16 rows-per-wave / 8 waves-per-block = 2048 blocks
    gate_wmma_kernel<<<2048, 256, 0, stream>>>(x, gw, g_ws);
    controls_kernel<<<TT * NN, TS, 0, stream>>>(g_ws, gb, out_ctrl, cs_ws, c1m_ws);
    forward_kernel<<<NN, 32, 0, stream>>>(em, cs_ws, c1m_ws, alpha_ws, prior_ws);
    backward_kernel<<<NN, 32, 0, stream>>>(em, cs_ws, c1m_ws, beta_ws);
    finalize_kernel<<<TT * NN, TS, 0, stream>>>(em, alpha_ws, beta_ws, prior_ws,
                                                cs_ws, c1m_ws, out);
}